// Attention_82617990906543
// MI455X (gfx1250) — compile-verified
//
#include <hip/hip_runtime.h>

typedef __attribute__((ext_vector_type(2))) float v2f;
typedef __attribute__((ext_vector_type(8))) float v8f;

#define B_  64
#define P_  8
#define N_  8192
#define D_  256
#define CH  16            // number of N-chunks (split-K over N)
#define NC  (N_ / CH)     // 512 keys per chunk
#define QS_STR 260        // padded LDS stride for Q (260*4 % 64banks -> conflict free)
#define SS_STR 516        // padded LDS stride for S
#define LOG_SCALE_MAX 4.6052f

// ---------------------------------------------------------------------------
// Kernel 1: per (batch, chunk) flash partial:
//   S = (Q*scale) K^T  -> chunk softmax -> O_part = P~ V, plus (m, l) stats.
// 8 waves / 256 threads per workgroup.
// ---------------------------------------------------------------------------
__global__ __launch_bounds__(256) void attn_partial(
    const float* __restrict__ q, const float* __restrict__ k,
    const float* __restrict__ v, const float* __restrict__ ls,
    float* __restrict__ wsO, float* __restrict__ wsM, float* __restrict__ wsL)
{
  __shared__ float Qs[16 * QS_STR];   // 16 rows (P padded) x 256, padded stride
  __shared__ float Ss[16 * SS_STR];   // 16 rows x 512 scores, padded stride
  __shared__ float scl[P_];

  const int bc   = blockIdx.x;
  const int b    = bc / CH;
  const int c    = bc % CH;
  const int n0   = c * NC;
  const int tid  = threadIdx.x;
  const int wave = tid >> 5;
  const int lane = tid & 31;
  const int lrow = lane & 15;
  const int lhi  = lane >> 4;

  if (tid < P_) scl[tid] = __expf(fminf(ls[tid], LOG_SCALE_MAX));
  __syncthreads();

  // Load Q scaled into LDS; pad rows 8..15 with zeros (so padded S rows are 0).
  for (int i = tid; i < 16 * D_; i += 256) {
    const int r = i >> 8, col = i & 255;
    float val = 0.f;
    if (r < P_) val = q[((size_t)b * P_ + r) * D_ + col] * scl[r];
    Qs[r * QS_STR + col] = val;
  }
  __syncthreads();

  // ---- Phase 1: S tiles via V_WMMA_F32_16X16X4_F32, K-depth = D = 256 ----
  for (int t = wave; t < NC / 16; t += 8) {
    // A fragment (Q): lane<16 holds Q[lrow][4k+0..1], lane>=16 holds Q[lrow][4k+2..3]
    const float* qbase = Qs + lrow * QS_STR + (lhi << 1);
    // B fragment (K^T): lane<16 holds K[n+lrow][4k+0..1], lane>=16 K[n+lrow][4k+2..3]
    const float* kbase = k + ((size_t)b * N_ + n0 + t * 16 + lrow) * D_ + (lhi << 1);
    v8f acc = {};
    #pragma unroll 8
    for (int kk = 0; kk < D_ / 4; ++kk) {
      v2f a  = *(const v2f*)(qbase + 4 * kk);
      v2f bb = *(const v2f*)(kbase + 4 * kk);
      acc = __builtin_amdgcn_wmma_f32_16x16x4_f32(false, a, false, bb,
                                                  (short)0, acc, false, false);
    }
    // C layout: VGPR r = row r (lanes 0-15) / row r+8 (lanes 16-31), col = lane%16
    #pragma unroll
    for (int r = 0; r < 8; ++r)
      Ss[((lhi << 3) + r) * SS_STR + t * 16 + lrow] = acc[r];
  }
  __syncthreads();

  // ---- Phase 2: chunk-local softmax, one wave per real query row ----
  {
    float* row = Ss + wave * SS_STR;          // waves 0..7 <-> rows 0..7
    float m = -3.4e38f;
    for (int i = lane; i < NC; i += 32) m = fmaxf(m, row[i]);
    #pragma unroll
    for (int o = 16; o > 0; o >>= 1) m = fmaxf(m, __shfl_xor(m, o, 32));
    float s = 0.f;
    for (int i = lane; i < NC; i += 32) {
      float e = __expf(row[i] - m);
      row[i] = e;
      s += e;
    }
    #pragma unroll
    for (int o = 16; o > 0; o >>= 1) s += __shfl_xor(s, o, 32);
    if (lane == 0) {
      const int idx = (b * CH + c) * P_ + wave;
      wsM[idx] = m;
      wsL[idx] = s;
    }
  }
  __syncthreads();

  // ---- Phase 3: O_part = P~ * V; each wave owns a 32-wide slice of D ----
  {
    const int d0 = wave * 32;
    // A fragment (P~ rows 0..15; padded rows are exact zeros)
    const float* abase = Ss + lrow * SS_STR + (lhi << 1);
    // B fragment (V): lane<16: V[n4+0..1][d0+lrow], lane>=16: V[n4+2..3][d0+lrow]
    const float* vb0 = v + ((size_t)b * N_ + n0 + (lhi << 1)) * D_ + d0 + lrow;
    const float* vb1 = vb0 + 16;
    v8f acc0 = {}, acc1 = {};
    #pragma unroll 4
    for (int kk = 0; kk < NC / 4; ++kk) {
      const size_t off = (size_t)kk * 4 * D_;
      v2f a = *(const v2f*)(abase + 4 * kk);
      v2f b0; b0.x = vb0[off]; b0.y = vb0[off + D_];
      v2f b1; b1.x = vb1[off]; b1.y = vb1[off + D_];
      acc0 = __builtin_amdgcn_wmma_f32_16x16x4_f32(false, a, false, b0,
                                                   (short)0, acc0, false, false);
      acc1 = __builtin_amdgcn_wmma_f32_16x16x4_f32(false, a, false, b1,
                                                   (short)0, acc1, false, false);
    }
    if (lane < 16) {  // rows 0..7 only (rows 8..15 are padding)
      float* outp = wsO + ((size_t)(b * CH + c) * P_) * D_ + d0 + lane;
      #pragma unroll
      for (int r = 0; r < 8; ++r) {
        outp[r * D_]      = acc0[r];
        outp[r * D_ + 16] = acc1[r];
      }
    }
  }
}

// ---------------------------------------------------------------------------
// Kernel 2: merge the CH chunk partials per batch (log-sum-exp combine).
// ---------------------------------------------------------------------------
__global__ __launch_bounds__(256) void attn_combine(
    const float* __restrict__ wsO, const float* __restrict__ wsM,
    const float* __restrict__ wsL, float* __restrict__ out)
{
  __shared__ float wgt[CH * P_];
  __shared__ float invL[P_];
  const int b   = blockIdx.x;
  const int tid = threadIdx.x;

  if (tid < P_) {
    float M = -3.4e38f;
    for (int c = 0; c < CH; ++c)
      M = fmaxf(M, wsM[(b * CH + c) * P_ + tid]);
    float L = 0.f;
    for (int c = 0; c < CH; ++c) {
      const float w = __expf(wsM[(b * CH + c) * P_ + tid] - M);
      wgt[c * P_ + tid] = w;
      L += w * wsL[(b * CH + c) * P_ + tid];
    }
    invL[tid] = 1.0f / L;
  }
  __syncthreads();

  for (int idx = tid; idx < P_ * D_; idx += 256) {
    const int p = idx >> 8, d = idx & 255;
    float acc = 0.f;
    for (int c = 0; c < CH; ++c)
      acc += wgt[c * P_ + p] * wsO[((size_t)(b * CH + c) * P_ + p) * D_ + d];
    out[((size_t)b * P_ + p) * D_ + d] = acc * invL[p];
  }
}

// ---------------------------------------------------------------------------
extern "C" void kernel_launch(void* const* d_in, const int* in_sizes, int n_in,
                              void* d_out, int out_size, void* d_ws, size_t ws_size,
                              hipStream_t stream) {
  (void)in_sizes; (void)n_in; (void)out_size; (void)ws_size;
  const float* q  = (const float*)d_in[0];
  const float* k  = (const float*)d_in[1];
  const float* v  = (const float*)d_in[2];
  const float* ls = (const float*)d_in[3];

  float* wsO = (float*)d_ws;                             // [B][CH][P][D]
  float* wsM = wsO + (size_t)B_ * CH * P_ * D_;          // [B][CH][P]
  float* wsL = wsM + (size_t)B_ * CH * P_;               // [B][CH][P]

  attn_partial<<<dim3(B_ * CH), dim3(256), 0, stream>>>(q, k, v, ls, wsO, wsM, wsL);
  attn_combine<<<dim3(B_), dim3(256), 0, stream>>>(wsO, wsM, wsL, (float*)d_out);
}